// S5Operator_32796370272331
// MI455X (gfx1250) — compile-verified
//
#include <hip/hip_runtime.h>
#include <math.h>

#define BSZ 8
#define H   1024
#define P   256
#define L   2048

typedef float v2f __attribute__((ext_vector_type(2)));
typedef float v8f __attribute__((ext_vector_type(8)));

// ---- workspace layout (float offsets) ----
#define OFF_LBAR 0                         // 2*P   : Lambda_bar (re,im interleaved)
#define OFF_BRE  (OFF_LBAR + 2*P)          // P*H   : Re(B_bar)[p][h]
#define OFF_BIM  (OFF_BRE + P*H)           // P*H   : Im(B_bar)[p][h]
#define OFF_CW   (OFF_BIM + P*H)           // 2P*H  : [2*C_re ; -2*C_im][p'][h]
#define OFF_BURE (OFF_CW + 2*P*H)          // BSZ*L*P : Re(Bu) -> Re(xs) in-place
#define OFF_BUIM (OFF_BURE + BSZ*L*P)      // BSZ*L*P : Im(Bu) -> Im(xs) in-place

// ------------------------------------------------------------------
// Kernel 0: ZOH discretization + operand prepack.
// One block per mode p.
// ------------------------------------------------------------------
__global__ __launch_bounds__(256) void s5_prep_kernel(
    const float* __restrict__ Lre, const float* __restrict__ Lim,
    const float* __restrict__ B,   const float* __restrict__ C,
    const float* __restrict__ logstep, float* __restrict__ ws)
{
    const int p = blockIdx.x;
    const float lre = Lre[p], lim = Lim[p];
    const float step = expf(logstep[p]);
    const float er  = expf(lre * step);
    const float lbr = er * cosf(lim * step);   // Lambda_bar (complex)
    const float lbi = er * sinf(lim * step);
    // coef = (Lambda_bar - 1) / Lambda   (complex division)
    const float denom = lre * lre + lim * lim;
    const float ar = lbr - 1.0f;
    const float cr = (ar * lre + lbi * lim) / denom;
    const float ci = (lbi * lre - ar * lim) / denom;

    if (threadIdx.x == 0) {
        ws[OFF_LBAR + 2 * p]     = lbr;
        ws[OFF_LBAR + 2 * p + 1] = lbi;
    }
    float* __restrict__ Bre = ws + OFF_BRE;
    float* __restrict__ Bim = ws + OFF_BIM;
    float* __restrict__ Cw  = ws + OFF_CW;
    for (int h = threadIdx.x; h < H; h += blockDim.x) {
        const float b0 = B[(size_t)(p * H + h) * 2];
        const float b1 = B[(size_t)(p * H + h) * 2 + 1];
        Bre[(size_t)p * H + h] = cr * b0 - ci * b1;   // Re(coef * B_tilde)
        Bim[(size_t)p * H + h] = cr * b1 + ci * b0;   // Im(coef * B_tilde)
        // C' rows: k = p -> 2*C_re ; k = p+P -> -2*C_im  (folds 2*Re(.))
        Cw[(size_t)p * H + h]       =  2.0f * C[(size_t)(h * P + p) * 2];
        Cw[(size_t)(p + P) * H + h] = -2.0f * C[(size_t)(h * P + p) * 2 + 1];
    }
}

// ------------------------------------------------------------------
// Kernel 1: Bu[b,l,p] = sum_h u[b,l,h] * B_bar[p,h]   (re & im)
// f32 WMMA 16x16x4. grid = (BSZ*L/16, P/128), block = 8 waves.
// Wave w owns p-tile p0 = by*128 + w*16 and produces both re & im
// accumulators sharing the A (u) fragment.
// ------------------------------------------------------------------
__global__ __launch_bounds__(256) void s5_gemm1_kernel(
    const float* __restrict__ x, float* __restrict__ ws)
{
    const int lane = threadIdx.x & 31;
    const int w    = threadIdx.x >> 5;
    const int mt   = blockIdx.x;
    const int b    = mt / (L / 16);
    const int l0   = (mt % (L / 16)) * 16;
    const int p0   = blockIdx.y * 128 + w * 16;

    const int m     = lane & 15;          // A row (within tile), also B col n
    const int khalf = (lane >> 4) * 2;    // K sub-offset for both A and B frags

    const float* __restrict__ Bre = ws + OFF_BRE;
    const float* __restrict__ Bim = ws + OFF_BIM;

    const float* xrow  = x + (size_t)b * H * L + l0 + m;       // + k*L
    const float* browR = Bre + (size_t)(p0 + m) * H;           // n == m bits
    const float* browI = Bim + (size_t)(p0 + m) * H;

    v8f accR = {0.f,0.f,0.f,0.f,0.f,0.f,0.f,0.f};
    v8f accI = {0.f,0.f,0.f,0.f,0.f,0.f,0.f,0.f};

    for (int k = 0; k < H; k += 4) {
        v2f a;  a.x  = xrow[(size_t)(k + khalf) * L];
                a.y  = xrow[(size_t)(k + khalf + 1) * L];
        v2f br; br.x = browR[k + khalf];  br.y = browR[k + khalf + 1];
        v2f bi; bi.x = browI[k + khalf];  bi.y = browI[k + khalf + 1];
        accR = __builtin_amdgcn_wmma_f32_16x16x4_f32(false, a, false, br,
                                                     (short)0, accR, false, false);
        accI = __builtin_amdgcn_wmma_f32_16x16x4_f32(false, a, false, bi,
                                                     (short)0, accI, false, false);
    }

    // C/D layout: lane n holds column p0+n, VGPR j -> row m = j (+8 for hi lanes)
    const int moff = (lane >> 4) * 8;
    float* __restrict__ buR = ws + OFF_BURE + ((size_t)(b * L + l0) * P) + p0 + m;
    float* __restrict__ buI = ws + OFF_BUIM + ((size_t)(b * L + l0) * P) + p0 + m;
#pragma unroll
    for (int j = 0; j < 8; ++j) {
        buR[(size_t)(moff + j) * P] = accR[j];
        buI[(size_t)(moff + j) * P] = accI[j];
    }
}

// ------------------------------------------------------------------
// Kernel 2: per-(b,p) complex scan  xs[l] = Lbar*xs[l-1] + Bu[l].
// One block per sequence; thread t scans 8 elements, block-wide
// Hillis-Steele scan of composed (A,b) pairs in LDS, then replay.
// ------------------------------------------------------------------
__global__ __launch_bounds__(256) void s5_scan_kernel(float* __restrict__ ws)
{
    const int bp = blockIdx.x;
    const int b  = bp >> 8;          // P == 256
    const int p  = bp & (P - 1);
    const float ar = ws[OFF_LBAR + 2 * p];
    const float ai = ws[OFF_LBAR + 2 * p + 1];
    const int t = threadIdx.x;

    float* __restrict__ buR = ws + OFF_BURE;
    float* __restrict__ buI = ws + OFF_BUIM;
    const size_t base = ((size_t)b * L + (size_t)t * 8) * P + p;

    float br[8], bi[8];
#pragma unroll
    for (int j = 0; j < 8; ++j) {
        br[j] = buR[base + (size_t)j * P];
        bi[j] = buI[base + (size_t)j * P];
    }

    // local scan from zero state
    float sr = 0.f, si = 0.f;
#pragma unroll
    for (int j = 0; j < 8; ++j) {
        const float nr = ar * sr - ai * si + br[j];
        const float ni = ar * si + ai * sr + bi[j];
        sr = nr; si = ni;
    }
    // A_chunk = Lbar^8 via repeated squaring
    const float a2r = ar * ar - ai * ai, a2i = 2.f * ar * ai;
    const float a4r = a2r * a2r - a2i * a2i, a4i = 2.f * a2r * a2i;
    const float a8r = a4r * a4r - a4i * a4i, a8i = 2.f * a4r * a4i;

    __shared__ float sAr[256], sAi[256], sBr[256], sBi[256];
    sAr[t] = a8r; sAi[t] = a8i; sBr[t] = sr; sBi[t] = si;
    __syncthreads();

    for (int off = 1; off < 256; off <<= 1) {
        float pAr = 0.f, pAi = 0.f, pBr = 0.f, pBi = 0.f;
        const bool act = (t >= off);
        if (act) { pAr = sAr[t-off]; pAi = sAi[t-off]; pBr = sBr[t-off]; pBi = sBi[t-off]; }
        __syncthreads();
        if (act) {
            const float cAr = sAr[t], cAi = sAi[t], cBr = sBr[t], cBi = sBi[t];
            sAr[t] = cAr * pAr - cAi * pAi;
            sAi[t] = cAr * pAi + cAi * pAr;
            sBr[t] = cAr * pBr - cAi * pBi + cBr;
            sBi[t] = cAr * pBi + cAi * pBr + cBi;
        }
        __syncthreads();
    }

    float crr = 0.f, cii = 0.f;
    if (t > 0) { crr = sBr[t - 1]; cii = sBi[t - 1]; }

    // replay with carry, write xs in place
    sr = crr; si = cii;
#pragma unroll
    for (int j = 0; j < 8; ++j) {
        const float nr = ar * sr - ai * si + br[j];
        const float ni = ar * si + ai * sr + bi[j];
        sr = nr; si = ni;
        buR[base + (size_t)j * P] = sr;
        buI[base + (size_t)j * P] = si;
    }
}

// ------------------------------------------------------------------
// Kernel 3: y[b,h,l] = gelu( [xs_re|xs_im]·C' + D[h]*u ), K = 2P = 512.
// grid = (BSZ*L/16, H/128). Transposed store: each lane owns 8
// contiguous l-elements -> two float4 stores.
// ------------------------------------------------------------------
__global__ __launch_bounds__(256) void s5_gemm2_kernel(
    const float* __restrict__ x, const float* __restrict__ Dv,
    const float* __restrict__ ws, float* __restrict__ out)
{
    const int lane = threadIdx.x & 31;
    const int w    = threadIdx.x >> 5;
    const int mt   = blockIdx.x;
    const int b    = mt / (L / 16);
    const int l0   = (mt % (L / 16)) * 16;
    const int h0   = blockIdx.y * 128 + w * 16;

    const int m     = lane & 15;
    const int khalf = (lane >> 4) * 2;

    const float* __restrict__ Are = ws + OFF_BURE + (size_t)(b * L + l0 + m) * P;
    const float* __restrict__ Aim = ws + OFF_BUIM + (size_t)(b * L + l0 + m) * P;
    const float* __restrict__ Cw  = ws + OFF_CW;

    v8f acc = {0.f,0.f,0.f,0.f,0.f,0.f,0.f,0.f};

    for (int k = 0; k < 2 * P; k += 4) {
        const float* Ak = (k < P) ? Are : (Aim - P);   // Ak[k+..] indexes correct half
        v2f a;  a.x  = Ak[k + khalf];
                a.y  = Ak[k + khalf + 1];
        v2f bb; bb.x = Cw[(size_t)(k + khalf) * H + h0 + m];
                bb.y = Cw[(size_t)(k + khalf + 1) * H + h0 + m];
        acc = __builtin_amdgcn_wmma_f32_16x16x4_f32(false, a, false, bb,
                                                    (short)0, acc, false, false);
    }

    // epilogue: feedthrough + exact GELU, transposed store out[b,h,l]
    const int   moff = (lane >> 4) * 8;
    const int   h    = h0 + m;                 // lane's column
    const float d    = Dv[h];
    const float* urow = x   + (size_t)b * H * L + (size_t)h * L + l0 + moff;
    float*       orow = out + (size_t)b * H * L + (size_t)h * L + l0 + moff;

    const float4 u0 = *(const float4*)(urow);
    const float4 u1 = *(const float4*)(urow + 4);
    const float uv[8] = {u0.x, u0.y, u0.z, u0.w, u1.x, u1.y, u1.z, u1.w};

    float res[8];
#pragma unroll
    for (int j = 0; j < 8; ++j) {
        const float v = acc[j] + d * uv[j];
        res[j] = 0.5f * v * (1.0f + erff(v * 0.70710678118654752f));
    }
    *(float4*)(orow)     = make_float4(res[0], res[1], res[2], res[3]);
    *(float4*)(orow + 4) = make_float4(res[4], res[5], res[6], res[7]);
}

// ------------------------------------------------------------------
extern "C" void kernel_launch(void* const* d_in, const int* in_sizes, int n_in,
                              void* d_out, int out_size, void* d_ws, size_t ws_size,
                              hipStream_t stream)
{
    const float* x       = (const float*)d_in[0];
    const float* Lre     = (const float*)d_in[1];
    const float* Lim     = (const float*)d_in[2];
    const float* B       = (const float*)d_in[3];
    const float* C       = (const float*)d_in[4];
    const float* Dv      = (const float*)d_in[5];
    const float* logstep = (const float*)d_in[6];
    float* ws  = (float*)d_ws;
    float* out = (float*)d_out;

    s5_prep_kernel<<<P, 256, 0, stream>>>(Lre, Lim, B, C, logstep, ws);

    dim3 g1(BSZ * L / 16, P / 128);
    s5_gemm1_kernel<<<g1, 256, 0, stream>>>(x, ws);

    s5_scan_kernel<<<BSZ * P, 256, 0, stream>>>(ws);

    dim3 g2(BSZ * L / 16, H / 128);
    s5_gemm2_kernel<<<g2, 256, 0, stream>>>(x, Dv, ws, out);
}